// LSTMLayer_77498389889243
// MI455X (gfx1250) — compile-verified
//
#include <hip/hip_runtime.h>
#include <math.h>

// LSTM (no forget gate), N=32 batch, L=2048 steps, C=128 in, O=512 out.
// z_t = [x_t, h] @ W + b with W = [Wf;Wr] in [640,1536].
// 16 persistent workgroups (1 per WGP), each owns 32 h-columns (96 gate cols).
// Weights live in LDS (240KB/WGP) for the whole kernel; fp32 WMMA 16x16x4.

typedef float v2f __attribute__((ext_vector_type(2)));
typedef float v8f __attribute__((ext_vector_type(8)));

#define NB   32      // batch
#define SEQ  2048
#define CIN  128
#define OUT  512
#define KTOT 640     // CIN + OUT
#define NWG  16      // workgroups / WGPs
#define COLS 96      // gate columns per WGP (32 per gate)
#define WGS  384     // 12 waves: 2 m-tiles x 6 n-tiles
#define KSTEPS 160   // KTOT / 4
#define BFLOATS (6 * KSTEPS * 64)   // 61440 swizzled B floats per WGP

__device__ __forceinline__ float sigmoidf_(float v) {
    return 1.0f / (1.0f + __expf(-v));
}

__global__ void __launch_bounds__(WGS, 1)
lstm_persistent_kernel(const float* __restrict__ x,
                       const float* __restrict__ Wf_ci, const float* __restrict__ Wf_ig,
                       const float* __restrict__ Wf_og,
                       const float* __restrict__ Wr_ci, const float* __restrict__ Wr_ig,
                       const float* __restrict__ Wr_og,
                       const float* __restrict__ b_ci, const float* __restrict__ b_ig,
                       const float* __restrict__ b_og,
                       const float* __restrict__ h_first, const float* __restrict__ c_first,
                       float* __restrict__ hbuf0, float* __restrict__ hbuf1,
                       unsigned* __restrict__ ctr,
                       float* __restrict__ out) {
    extern __shared__ float lds[];
    float* Bs   = lds;                   // 61440 floats: swizzled weight shard
    float* zbuf = lds + BFLOATS;         // 3072 floats: z staging [32 x 96]
    float* cst  = lds + BFLOATS + 3072;  // 1024 floats: cell state [32 x 32]

    const int tid  = threadIdx.x;
    const int wg   = blockIdx.x;     // 0..15 -> h columns [wg*32, wg*32+32)
    const int wave = tid >> 5;
    const int lane = tid & 31;
    const int mt   = wave & 1;       // m-tile (rows mt*16..mt*16+15)
    const int nt   = wave >> 1;      // n-tile 0..5 (0,1=ci 2,3=ig 4,5=og)
    const int half = lane >> 4;
    const int l16  = lane & 15;

    const float* WfG[3] = {Wf_ci, Wf_ig, Wf_og};
    const float* WrG[3] = {Wr_ci, Wr_ig, Wr_og};
    const float* bG[3]  = {b_ci,  b_ig,  b_og };

    // ---- Stage weight shard into LDS, pre-swizzled into WMMA B-fragment order.
    // Fragment for (ntile, kstep): 64 floats, index = lane*2 + v, holding
    // B[K = 4k + v + 2*(lane>>4)][col = wg*32 + (nt&1)*16 + (lane&15)].
    for (int idx = tid; idx < BFLOATS; idx += WGS) {
        int blk    = idx >> 6;
        int within = idx & 63;
        int ln = within >> 1, v = within & 1;
        int bnt = blk / KSTEPS, k = blk % KSTEPS;
        int bh = ln >> 4, bl = ln & 15;
        int K    = 4 * k + v + 2 * bh;
        int gate = bnt >> 1;
        int col  = wg * 32 + (bnt & 1) * 16 + bl;
        float w = (K < CIN) ? WfG[gate][K * OUT + col]
                            : WrG[gate][(K - CIN) * OUT + col];
        Bs[idx] = w;
    }
    // ---- Init cell state + our shard of h0.
    for (int e = tid; e < NB * 32; e += WGS) {
        int m = e >> 5, n = e & 31;
        int col = wg * 32 + n;
        cst[e] = c_first[col];
        hbuf0[m * OUT + col] = h_first[col];
    }
    __threadfence();
    __syncthreads();
    // ---- Grid barrier #0: everyone's h0 shard visible before step 0.
    if (tid == 0) {
        atomicAdd(ctr, 1u);
        while (atomicAdd(ctr, 0u) < (unsigned)NWG) { __builtin_amdgcn_s_sleep(1); }
    }
    __syncthreads();
    __threadfence();

    // Per-lane bias for this wave's output column (baked into the accumulator).
    const float bval = bG[nt >> 1][wg * 32 + (nt & 1) * 16 + l16];

    const int mrow = mt * 16 + l16;                 // A-fragment batch row
    const float* xrow = x + (size_t)mrow * (SEQ * CIN);
    const float* bsp  = Bs + nt * (KSTEPS * 64) + lane * 2;

    for (int t = 0; t < SEQ; ++t) {
        const float* hr = (t & 1) ? hbuf1 : hbuf0;  // h_{t}
        float*       hw = (t & 1) ? hbuf0 : hbuf1;  // h_{t+1}

        v8f acc;
        #pragma unroll
        for (int r = 0; r < 8; ++r) acc[r] = bval;

        // K-part 1: x_t  (K = 0..127)
        const float* ap = xrow + t * CIN + 2 * half;
        #pragma unroll 4
        for (int k = 0; k < CIN / 4; ++k) {
            v2f a  = *(const v2f*)(ap + 4 * k);
            v2f bb = *(const v2f*)(bsp + 64 * k);
            acc = __builtin_amdgcn_wmma_f32_16x16x4_f32(
                false, a, false, bb, (short)0, acc, false, false);
        }
        // K-part 2: h_t  (K = 128..639)
        const float* hp   = hr + mrow * OUT + 2 * half;
        const float* bsp2 = bsp + (CIN / 4) * 64;
        #pragma unroll 4
        for (int k = 0; k < OUT / 4; ++k) {
            v2f a  = *(const v2f*)(hp + 4 * k);
            v2f bb = *(const v2f*)(bsp2 + 64 * k);
            acc = __builtin_amdgcn_wmma_f32_16x16x4_f32(
                false, a, false, bb, (short)0, acc, false, false);
        }
        // D layout: VGPR r -> (row r | row r+8), col = lane&15. Stage to LDS.
        #pragma unroll
        for (int r = 0; r < 8; ++r) {
            int row = r + 8 * half;
            zbuf[row * COLS + nt * 16 + l16] = acc[r];
        }
        __syncthreads();

        // Prefetch next step's x tile while gates run.
        if (t + 1 < SEQ && tid < 128) {
            int pr = tid >> 2, seg = tid & 3;
            __builtin_prefetch(x + ((size_t)pr * SEQ + (t + 1)) * CIN + seg * 32, 0, 0);
        }

        // Gates + state update for our 32x32 h shard.
        const bool last = (t == SEQ - 1);
        for (int e = tid; e < NB * 32; e += WGS) {
            int m = e >> 5, n = e & 31;
            float zci = zbuf[m * COLS + n];
            float zig = zbuf[m * COLS + 32 + n];
            float zog = zbuf[m * COLS + 64 + n];
            float ci = tanhf(zci);
            float ig = sigmoidf_(zig);
            float og = sigmoidf_(zog);
            float c  = cst[e] + ci * ig;           // forget gate == 1
            cst[e] = c;
            float h = tanhf(c) * og;
            hw[m * OUT + wg * 32 + n] = h;
            if (last) out[m * OUT + wg * 32 + n] = h;
        }
        __threadfence();          // release our h shard to device scope
        __syncthreads();
        // ---- Grid barrier: monotonic counter, one signal per workgroup.
        if (tid == 0) {
            atomicAdd(ctr, 1u);
            unsigned target = (unsigned)NWG * (unsigned)(t + 2);
            while (atomicAdd(ctr, 0u) < target) { __builtin_amdgcn_s_sleep(1); }
        }
        __syncthreads();
        __threadfence();          // acquire: invalidate stale h lines in WGP$
    }
}

extern "C" void kernel_launch(void* const* d_in, const int* in_sizes, int n_in,
                              void* d_out, int out_size, void* d_ws, size_t ws_size,
                              hipStream_t stream) {
    const float* x     = (const float*)d_in[0];
    const float* Wf_ci = (const float*)d_in[1];
    const float* Wf_ig = (const float*)d_in[2];
    const float* Wf_og = (const float*)d_in[3];
    const float* Wr_ci = (const float*)d_in[4];
    const float* Wr_ig = (const float*)d_in[5];
    const float* Wr_og = (const float*)d_in[6];
    const float* b_ci  = (const float*)d_in[7];
    const float* b_ig  = (const float*)d_in[8];
    const float* b_og  = (const float*)d_in[9];
    const float* h_first = (const float*)d_in[10];
    const float* c_first = (const float*)d_in[11];
    float* out = (float*)d_out;

    float* hbuf0 = (float*)d_ws;                 // [32 x 512]
    float* hbuf1 = hbuf0 + NB * OUT;             // [32 x 512]
    unsigned* ctr = (unsigned*)(hbuf1 + NB * OUT);

    // Zero h double-buffer + barrier counter (graph-capture-safe stream op).
    hipMemsetAsync(d_ws, 0, (size_t)(2 * NB * OUT) * sizeof(float) + 64, stream);

    const size_t lds_bytes = (size_t)(BFLOATS + 3072 + 1024) * sizeof(float); // 256 KB
    lstm_persistent_kernel<<<NWG, WGS, lds_bytes, stream>>>(
        x, Wf_ci, Wf_ig, Wf_og, Wr_ci, Wr_ig, Wr_og,
        b_ci, b_ig, b_og, h_first, c_first,
        hbuf0, hbuf1, ctr, out);
}